// RelativePositionEncoding_81372450390562
// MI455X (gfx1250) — compile-verified
//
#include <hip/hip_runtime.h>

#ifndef __has_builtin
#define __has_builtin(x) 0
#endif

#define R_MAX   32
#define S_MAX   2
#define R_BINS  66            // 2*R_MAX + 2
#define S_BINS  6             // 2*S_MAX + 2
#define DIM_IN  139           // 2*R_BINS + 1 + S_BINS
#define DIM_OUT 128

typedef __attribute__((ext_vector_type(4))) float f4;
typedef int i4 __attribute__((vector_size(16)));              // matches builtin proto
typedef __attribute__((address_space(1))) i4 gi4;             // global int4
typedef __attribute__((address_space(3))) i4 li4;             // LDS int4

__device__ __forceinline__ void wait_async_all() {
#if __has_builtin(__builtin_amdgcn_s_wait_asynccnt)
    __builtin_amdgcn_s_wait_asynccnt(0);
#else
    asm volatile("s_wait_asynccnt 0" ::: "memory");
#endif
}

__global__ __launch_bounds__(256)
void relpos_kernel(const float* __restrict__ feats,   // [1, n, 10]
                   const float* __restrict__ W,       // [139, 128]
                   float* __restrict__ out,           // [1, n, n, 128]
                   int n)
{
    __shared__ float sW[DIM_IN * DIM_OUT];   // 71,168 B of the 320 KB WGP LDS
    __shared__ int   sF[2][16][8];           // [i-side/j-side][row][5 feats, padded]

    const int tid  = threadIdx.x;
    const int lane = tid & 31;
    const int wave = tid >> 5;               // 8 wave32s per block

    const int ti = blockIdx.y << 4;          // i-tile base (2D grid: no div/mod)
    const int tj = blockIdx.x << 4;          // j-tile base

    // ---- stage W into LDS via CDNA5 async copy engine (ASYNCcnt-tracked) ----
    const int NW4 = (DIM_IN * DIM_OUT) / 4;  // 4448 float4s
#if __has_builtin(__builtin_amdgcn_global_load_async_to_lds_b128)
    for (int v = tid; v < NW4; v += 256) {
        gi4* g1 = (gi4*)(void*)(W + (size_t)v * 4);
        li4* l3 = (li4*)(void*)(&sW[v * 4]);
        __builtin_amdgcn_global_load_async_to_lds_b128(g1, l3, 0, 0);
    }
#else
    for (int v = tid; v < NW4; v += 256) {
        ((f4*)sW)[v] = ((const f4*)W)[v];
    }
#endif

    // ---- stage the 32 needed tokens' integer features (values fit exactly) ----
    if (tid < 32) {
        const int which = tid >> 4;              // 0 = i side, 1 = j side
        const int r     = tid & 15;
        const int tok   = (which ? tj : ti) + r;
        const float* fp = feats + (size_t)tok * 10;
        #pragma unroll
        for (int c = 0; c < 5; ++c)
            sF[which][r][c] = (int)fp[c];
    }

#if __has_builtin(__builtin_amdgcn_global_load_async_to_lds_b128)
    wait_async_all();
#endif
    __syncthreads();

    const int col  = lane << 2;                             // 4 floats per lane
    const f4  w132 = *(const f4*)&sW[132 * DIM_OUT + col];  // entity-mask row, hoisted

    // Per-wave output base: pairs q = it*8 + wave, io = q>>4, jo = q&15.
    float* const obase =
        out + (((size_t)ti * n + tj) * DIM_OUT + col);

    // 256 pairs per block, one pair per wave per iteration.
    for (int it = 0; it < 32; ++it) {
        const int q  = (it << 3) + wave;               // 0..255
        const int io = q >> 4;
        const int jo = q & 15;

        const int ri = sF[0][io][0], ki = sF[0][io][1], ai = sF[0][io][2],
                  ei = sF[0][io][3], si = sF[0][io][4];
        const int rj = sF[1][jo][0], kj = sF[1][jo][1], aj = sF[1][jo][2],
                  ej = sF[1][jo][3], sj = sF[1][jo][4];

        const int  dr  = ri - rj;
        const int  dk  = ki - kj;
        const int  dsy = si - sj;
        const bool same_chain = (ai == aj);
        const bool same_res   = (dr == 0);
        const float m = (ei == ej) ? 1.0f : 0.0f;

        const int d_res = same_chain ? min(max(dr + R_MAX, 0), 2 * R_MAX)
                                     : (2 * R_MAX + 1);
        const int d_tok = (same_chain && same_res) ? min(max(dk + R_MAX, 0), 2 * R_MAX)
                                                   : (2 * R_MAX + 1);
        const int d_chn = same_chain ? (2 * S_MAX + 1)
                                     : min(max(dsy + S_MAX, 0), 2 * S_MAX);

        const f4 w0 = *(const f4*)&sW[d_res * DIM_OUT + col];
        const f4 w1 = *(const f4*)&sW[(R_BINS + d_tok) * DIM_OUT + col];
        const f4 w3 = *(const f4*)&sW[(2 * R_BINS + 1 + d_chn) * DIM_OUT + col];

        f4 r;
        r.x = fmaf(m, w132.x, w0.x + w1.x + w3.x);
        r.y = fmaf(m, w132.y, w0.y + w1.y + w3.y);
        r.z = fmaf(m, w132.z, w0.z + w1.z + w3.z);
        r.w = fmaf(m, w132.w, w0.w + w1.w + w3.w);

        f4* op = (f4*)(obase + ((size_t)io * n + jo) * DIM_OUT);
        __builtin_nontemporal_store(r, op);   // 512 MiB write-once: stream past L2
    }
}

extern "C" void kernel_launch(void* const* d_in, const int* in_sizes, int n_in,
                              void* d_out, int out_size, void* d_ws, size_t ws_size,
                              hipStream_t stream)
{
    const float* feats = (const float*)d_in[0];   // [1, n, 10] f32
    const float* W     = (const float*)d_in[1];   // [139, 128] f32
    float* out = (float*)d_out;                   // [1, n, n, 128] f32
    const int n = in_sizes[0] / 10;               // n = 1024
    dim3 grid(n >> 4, n >> 4);                    // (j-tiles, i-tiles)
    relpos_kernel<<<grid, 256, 0, stream>>>(feats, W, out, n);
}